// SecondGCN_1958505087038
// MI455X (gfx1250) — compile-verified
//
#include <hip/hip_runtime.h>
#include <hip/hip_bf16.h>
#include <cstdint>

// ---------------------------------------------------------------------------
// 3-layer GCN for MI455X (gfx1250, wave32).
//  - Dense projections: V_WMMA_F32_16X16X4_F32, W staged zero-padded in LDS
//    (branch-free inner loop: global_load_b64 for A, ds b64 loads for B,
//     B loads batched so one s_wait_dscnt covers all NTILES fetches).
//  - Edge aggregation: wave-per-edge float2 gather + global_atomic_add_f32.
//    Feature matrices (100K x 50 = 20MB) fit in the 192MB L2, so the random
//    gather/scatter runs at L2 bandwidth; HBM only streams edges + x once.
// ---------------------------------------------------------------------------

typedef float v2f __attribute__((ext_vector_type(2)));
typedef float v8f __attribute__((ext_vector_type(8)));

#define WAVE 32

// ---- degree / normalization -----------------------------------------------

__global__ void deg_init_kernel(float* __restrict__ deg, int n) {
  int i = blockIdx.x * blockDim.x + threadIdx.x;
  if (i < n) deg[i] = 1.0f;  // self-loop contributes weight 1
}

__global__ void deg_accum_kernel(const int* __restrict__ col,
                                 const float* __restrict__ ew,
                                 float* __restrict__ deg, int e) {
  int i = blockIdx.x * blockDim.x + threadIdx.x;
  if (i < e) unsafeAtomicAdd(&deg[col[i]], ew[i]);
}

__global__ void dinv_kernel(float* __restrict__ deg, int n) {
  int i = blockIdx.x * blockDim.x + threadIdx.x;
  if (i < n) {
    float d = deg[i];
    deg[i] = (d > 0.0f) ? rsqrtf(d) : 0.0f;  // in-place deg -> D^-1/2
  }
}

// ---- dense projection: C[M x Nout] = A[M x K] @ W[K x Nout], fp32 WMMA ----
// One wave computes a 16 x (NTILES*16) strip (one WMMA 16x16 tile per nt).
// W is staged in LDS zero-padded to [NTILES*16][Kpad] with row stride S
// (S % 64 == 4 -> the b64 read pattern n*S + {kb,kb+1} is bank-conflict-free).
// Since B == 0 for k >= K, A may be loaded unguarded (buffers carry slack).
// A 16x4 layout : lane l -> row m = l%16, ks {kb, kb+1}, kb = 2*(l/16)
// B 4x16 layout : lane l -> col n = l%16, ks {kb, kb+1}
// C/D layout    : VGPR r, lane l -> row r + 8*(l/16), col l%16
template <int NTILES>
__global__ void __launch_bounds__(128)
gemm_wmma_f32(const float* __restrict__ A, const float* __restrict__ W,
              float* __restrict__ C, int M, int K, int Kpad, int S, int Nout) {
  extern __shared__ float Wlds[];  // NTILES*16 * S floats

  // --- cooperative zero-padded fill of W into LDS (once per block) ---
  const int Npad = NTILES * 16;
  for (int t = threadIdx.x; t < Npad * Kpad; t += blockDim.x) {
    int n = t / Kpad;
    int k = t - n * Kpad;
    float v = (n < Nout && k < K) ? W[(size_t)k * Nout + n] : 0.0f;
    Wlds[n * S + k] = v;
  }
  __syncthreads();

  int mt = blockIdx.x * (blockDim.x / WAVE) + (threadIdx.x / WAVE);
  if (mt * 16 >= M) return;  // wave-uniform: EXEC stays all-1s for WMMA

  int lane = threadIdx.x & (WAVE - 1);
  int mloc = lane & 15;
  int kb   = (lane >> 4) << 1;
  const float* __restrict__ Arow = A + (size_t)(mt * 16 + mloc) * K;

  // per-lane LDS base pointer for each n-tile's column (hoisted)
  const float* __restrict__ Bbase[NTILES];
#pragma unroll
  for (int nt = 0; nt < NTILES; ++nt)
    Bbase[nt] = &Wlds[(nt * 16 + mloc) * S + kb];

  v8f acc[NTILES];
#pragma unroll
  for (int nt = 0; nt < NTILES; ++nt)
    acc[nt] = (v8f){0.f, 0.f, 0.f, 0.f, 0.f, 0.f, 0.f, 0.f};

  for (int k0 = 0; k0 < Kpad; k0 += 4) {
    v2f a = *(const v2f*)(Arow + k0 + kb);  // 8B aligned: K even, k0+kb even
    // batch all B loads first (distinct regs -> single dscnt wait),
    // then issue the NTILES WMMAs back-to-back (independent accumulators).
    v2f b[NTILES];
#pragma unroll
    for (int nt = 0; nt < NTILES; ++nt)
      b[nt] = *(const v2f*)(Bbase[nt] + k0);
#pragma unroll
    for (int nt = 0; nt < NTILES; ++nt)
      // (neg_a, A, neg_b, B, c_mod, C, reuse_a, reuse_b)
      acc[nt] = __builtin_amdgcn_wmma_f32_16x16x4_f32(false, a, false, b[nt],
                                                      (short)0, acc[nt],
                                                      false, false);
  }

  int r0 = mt * 16 + ((lane >> 4) << 3);
#pragma unroll
  for (int nt = 0; nt < NTILES; ++nt) {
    int nc = nt * 16 + mloc;
    if (nc < Nout) {
      float* __restrict__ Crow = C + (size_t)r0 * Nout + nc;
#pragma unroll
      for (int r = 0; r < 8; ++r) Crow[(size_t)r * Nout] = acc[nt][r];
    }
  }
}

// ---- aggregation -----------------------------------------------------------

// out[i,:] = dinv[i]^2 * t[i,:]   (self-loop term; also initializes out)
__global__ void selfloop_init_kernel(const float* __restrict__ t,
                                     const float* __restrict__ dinv,
                                     float* __restrict__ out, int F) {
  int i = blockIdx.x;
  float d  = dinv[i];
  float dd = d * d;
  const float* __restrict__ src = t + (size_t)i * F;
  float* __restrict__ dst = out + (size_t)i * F;
  for (int f = threadIdx.x; f < F; f += blockDim.x) dst[f] = dd * src[f];
}

// out[col[e],:] += (dinv[row]*ew*dinv[col]) * t[row[e],:]
// F even; EPW edges per wave, each edge served by F/2 lanes doing float2.
template <int F>
__global__ void edge_scatter_kernel(const float* __restrict__ t,
                                    const int* __restrict__ row,
                                    const int* __restrict__ col,
                                    const float* __restrict__ ew,
                                    const float* __restrict__ dinv,
                                    float* __restrict__ out, int E) {
  constexpr int PAIRS  = F / 2;                          // float2s per row
  constexpr int EPW    = (32 / PAIRS) > 0 ? (32 / PAIRS) : 1;
  constexpr int ACTIVE = PAIRS * EPW;
  int wv   = blockIdx.x * (blockDim.x / WAVE) + (threadIdx.x / WAVE);
  int lane = threadIdx.x & (WAVE - 1);
  int sub  = lane / PAIRS;
  int p    = lane - sub * PAIRS;
  int e    = wv * EPW + sub;
  if (lane >= ACTIVE || e >= E) return;
  int r = row[e];
  int c = col[e];
  float w = dinv[r] * ew[e] * dinv[c];
  float2 v = ((const float2*)(t + (size_t)r * F))[p];    // 8B aligned (F even)
  float* dst = out + (size_t)c * F + 2 * p;
  unsafeAtomicAdd(dst, v.x * w);
  unsafeAtomicAdd(dst + 1, v.y * w);
}

// generic fallback (any F)
__global__ void edge_scatter_generic(const float* __restrict__ t,
                                     const int* __restrict__ row,
                                     const int* __restrict__ col,
                                     const float* __restrict__ ew,
                                     const float* __restrict__ dinv,
                                     float* __restrict__ out, int E, int F) {
  int e = blockIdx.x * (blockDim.x / WAVE) + (threadIdx.x / WAVE);
  if (e >= E) return;
  int lane = threadIdx.x & (WAVE - 1);
  int r = row[e];
  int c = col[e];
  float w = dinv[r] * ew[e] * dinv[c];
  const float* __restrict__ src = t + (size_t)r * F;
  float* __restrict__ dst = out + (size_t)c * F;
  for (int f = lane; f < F; f += WAVE) unsafeAtomicAdd(&dst[f], src[f] * w);
}

// h[i,:] = act(h[i,:] + b)
__global__ void bias_act_kernel(float* __restrict__ h, const float* __restrict__ b,
                                int F, int relu) {
  int i = blockIdx.x;
  float* __restrict__ p = h + (size_t)i * F;
  for (int f = threadIdx.x; f < F; f += blockDim.x) {
    float v = p[f] + b[f];
    p[f] = relu ? fmaxf(v, 0.0f) : v;
  }
}

// ---------------------------------------------------------------------------

extern "C" void kernel_launch(void* const* d_in, const int* in_sizes, int n_in,
                              void* d_out, int out_size, void* d_ws, size_t ws_size,
                              hipStream_t stream) {
  const float* x  = (const float*)d_in[0];
  const int*   ei = (const int*)d_in[1];
  const float* ew = (const float*)d_in[2];
  const float* W1 = (const float*)d_in[3];
  const float* b1 = (const float*)d_in[4];
  const float* W2 = (const float*)d_in[5];
  const float* b2 = (const float*)d_in[6];
  const float* W3 = (const float*)d_in[7];
  const float* b3 = (const float*)d_in[8];
  float* out = (float*)d_out;

  const int H   = in_sizes[4];        // 50
  const int Cc  = in_sizes[8];        // 10
  const int FIN = in_sizes[3] / H;    // 128
  const int N   = in_sizes[0] / FIN;  // 100000
  const int E   = in_sizes[2];        // 3200000
  const int* row = ei;                // edge_index[0,:]
  const int* col = ei + E;            // edge_index[1,:]

  // ws layout (+16 floats slack per buffer: A operand may over-read <=2
  // elements past the last row when Kpad > K; the zero B rows cancel them).
  float* ws   = (float*)d_ws;
  float* dinv = ws;                          // N
  float* bufA = dinv + N;                    // N*H + 16
  float* bufB = bufA + (size_t)N * H + 16;   // N*H + 16
  float* bufC = bufB + (size_t)N * H + 16;   // N*H + 16

  // --- symmetric normalization: deg = 1 + sum_e ew[col], dinv = deg^-1/2 ---
  deg_init_kernel<<<(N + 255) / 256, 256, 0, stream>>>(dinv, N);
  deg_accum_kernel<<<(E + 255) / 256, 256, 0, stream>>>(col, ew, dinv, E);
  dinv_kernel<<<(N + 255) / 256, 256, 0, stream>>>(dinv, N);

  auto gemm = [&](const float* Ain, const float* Wm, float* Cout, int K, int Nout) {
    int Kpad = (K + 3) & ~3;
    int S    = ((Kpad + 63) & ~63) + 4;  // LDS row stride, S % 64 == 4
    int mTiles = (N + 15) / 16;
    int wavesPerBlock = 4;  // 128 threads
    int blocks = (mTiles + wavesPerBlock - 1) / wavesPerBlock;
    int nTiles = (Nout + 15) / 16;
    if (nTiles >= 4) {
      size_t ldsBytes = (size_t)4 * 16 * S * sizeof(float);
      gemm_wmma_f32<4><<<blocks, wavesPerBlock * WAVE, ldsBytes, stream>>>(
          Ain, Wm, Cout, N, K, Kpad, S, Nout);
    } else if (nTiles == 2) {
      size_t ldsBytes = (size_t)2 * 16 * S * sizeof(float);
      gemm_wmma_f32<2><<<blocks, wavesPerBlock * WAVE, ldsBytes, stream>>>(
          Ain, Wm, Cout, N, K, Kpad, S, Nout);
    } else {
      size_t ldsBytes = (size_t)1 * 16 * S * sizeof(float);
      gemm_wmma_f32<1><<<blocks, wavesPerBlock * WAVE, ldsBytes, stream>>>(
          Ain, Wm, Cout, N, K, Kpad, S, Nout);
    }
  };

  auto aggregate = [&](const float* t, float* agg, const float* bias, int F, int relu) {
    selfloop_init_kernel<<<N, 64, 0, stream>>>(t, dinv, agg, F);
    const int wpb = 8;  // 256 threads = 8 waves
    if (F == 50) {
      int epb = wpb * 1;  // EPW = 1
      edge_scatter_kernel<50><<<(E + epb - 1) / epb, wpb * WAVE, 0, stream>>>(
          t, row, col, ew, dinv, agg, E);
    } else if (F == 10) {
      int epb = wpb * 6;  // EPW = 6
      edge_scatter_kernel<10><<<(E + epb - 1) / epb, wpb * WAVE, 0, stream>>>(
          t, row, col, ew, dinv, agg, E);
    } else {
      edge_scatter_generic<<<(E + wpb - 1) / wpb, wpb * WAVE, 0, stream>>>(
          t, row, col, ew, dinv, agg, E, F);
    }
    bias_act_kernel<<<N, 64, 0, stream>>>(agg, bias, F, relu);
  };

  // Layer 1: 128 -> 50, ReLU
  gemm(x, W1, bufA, FIN, H);
  aggregate(bufA, bufB, b1, H, 1);
  // Layer 2: 50 -> 50, ReLU
  gemm(bufB, W2, bufA, H, H);
  aggregate(bufA, bufC, b2, H, 1);
  // Layer 3: 50 -> 10, no activation, straight into d_out
  gemm(bufC, W3, bufA, H, Cc);
  aggregate(bufA, out, b3, Cc, 0);
}